// GATLayer_27590869909900
// MI455X (gfx1250) — compile-verified
//
#include <hip/hip_runtime.h>

#define N_NODES 100000
#define N_EDGES 1600000
#define IN_DIM  256
#define OUT_DIM 64
#define N_TILES (N_NODES / 16)   // 6250, exact

typedef __attribute__((ext_vector_type(2))) float v2f;
typedef __attribute__((ext_vector_type(8))) float v8f;

// ---------------------------------------------------------------------------
// Zero-init: h_N output accumulator, segment-max m, segment-sum ssum
// ---------------------------------------------------------------------------
__global__ void k_zero(float* __restrict__ hN, float* __restrict__ m,
                       float* __restrict__ ssum) {
    const int stride = gridDim.x * blockDim.x;
    int i = blockIdx.x * blockDim.x + threadIdx.x;
    for (int j = i; j < N_NODES * OUT_DIM; j += stride) hN[j] = 0.0f;
    for (int j = i; j < N_NODES; j += stride) { m[j] = 0.0f; ssum[j] = 0.0f; }
}

// ---------------------------------------------------------------------------
// W_h = h @ W^T via V_WMMA_F32_16X16X4_F32, fused with per-node attention
// scalars s_src[n] = W_h[n] . a[0:64], s_dst[n] = W_h[n] . a[64:128].
// One wave computes a 16-row x 64-col tile (4 accumulators), K-loop step 4.
// ---------------------------------------------------------------------------
__global__ void k_gemm(const float* __restrict__ h, const float* __restrict__ W,
                       const float* __restrict__ a, float* __restrict__ Wh,
                       float* __restrict__ s_src, float* __restrict__ s_dst) {
    const int lane = threadIdx.x & 31;
    const int wave = threadIdx.x >> 5;
    const int tile = blockIdx.x * (blockDim.x >> 5) + wave;
    if (tile >= N_TILES) return;                 // wave-uniform: EXEC all-1s inside

    const int row0  = tile * 16;
    const int mrow  = lane & 15;                 // A: M index / B,C: N index (low half)
    const int khalf = (lane >> 4) << 1;          // 0 for lanes 0-15, 2 for lanes 16-31

    v8f c0 = {}, c1 = {}, c2 = {}, c3 = {};
    const float* ha = h + (size_t)(row0 + mrow) * IN_DIM + khalf;
    const float* wb = W + (size_t)mrow * IN_DIM + khalf;   // N-tile 0 column base

#pragma unroll 4
    for (int kk = 0; kk < IN_DIM; kk += 4) {
        v2f av = *(const v2f*)(ha + kk);
        v2f b0 = *(const v2f*)(wb + kk);
        v2f b1 = *(const v2f*)(wb + 16 * IN_DIM + kk);
        v2f b2 = *(const v2f*)(wb + 32 * IN_DIM + kk);
        v2f b3 = *(const v2f*)(wb + 48 * IN_DIM + kk);
        // 8 args: (neg_a, A, neg_b, B, c_mod, C, reuse_a, reuse_b)
        c0 = __builtin_amdgcn_wmma_f32_16x16x4_f32(false, av, false, b0, (short)0, c0, false, false);
        c1 = __builtin_amdgcn_wmma_f32_16x16x4_f32(false, av, false, b1, (short)0, c1, false, false);
        c2 = __builtin_amdgcn_wmma_f32_16x16x4_f32(false, av, false, b2, (short)0, c2, false, false);
        c3 = __builtin_amdgcn_wmma_f32_16x16x4_f32(false, av, false, b3, (short)0, c3, false, false);
    }

    // C/D layout: VGPR r -> lanes 0-15: (M=r, N=lane); lanes 16-31: (M=r+8, N=lane-16)
    const int colL   = lane & 15;
    const int rowOff = (lane >> 4) * 8;
    const float as0 = a[colL],      as1 = a[16 + colL],
                as2 = a[32 + colL], as3 = a[48 + colL];
    const float ad0 = a[64 + colL],  ad1 = a[80 + colL],
                ad2 = a[96 + colL],  ad3 = a[112 + colL];

#pragma unroll
    for (int r = 0; r < 8; ++r) {
        const int row = row0 + rowOff + r;
        const float w0 = c0[r], w1 = c1[r], w2 = c2[r], w3 = c3[r];
        float* o = Wh + (size_t)row * OUT_DIM + colL;
        o[0] = w0; o[16] = w1; o[32] = w2; o[48] = w3;

        float vs = w0 * as0 + w1 * as1 + w2 * as2 + w3 * as3;
        float vd = w0 * ad0 + w1 * ad1 + w2 * ad2 + w3 * ad3;
        // reduce across the 16 lanes that share this row (xor stays in-half)
#pragma unroll
        for (int off = 8; off >= 1; off >>= 1) {
            vs += __shfl_xor(vs, off, 32);
            vd += __shfl_xor(vd, off, 32);
        }
        if (colL == 0) { s_src[row] = vs; s_dst[row] = vd; }
    }
}

// ---------------------------------------------------------------------------
// e[k] = relu(s_src[src[k]] + s_dst[dst[k]]);  m[dst] = segment max (uint trick,
// valid since e >= 0 and empty segments never feed the output).
// ---------------------------------------------------------------------------
__global__ void k_edge_e(const int* __restrict__ src, const int* __restrict__ dst,
                         const float* __restrict__ s_src, const float* __restrict__ s_dst,
                         float* __restrict__ e, unsigned* __restrict__ m) {
    const int k = blockIdx.x * blockDim.x + threadIdx.x;
    if (k >= N_EDGES) return;
    const int d = dst[k];
    float v = s_src[src[k]] + s_dst[d];
    v = v > 0.0f ? v : 0.0f;
    e[k] = v;
    atomicMax(m + d, __float_as_uint(v));
}

// ---------------------------------------------------------------------------
// ex[k] = exp(e[k] - m[dst[k]]) (in place); ssum[dst] += ex
// ---------------------------------------------------------------------------
__global__ void k_edge_exp(const int* __restrict__ dst, const float* __restrict__ m,
                           float* __restrict__ e, float* __restrict__ ssum) {
    const int k = blockIdx.x * blockDim.x + threadIdx.x;
    if (k >= N_EDGES) return;
    const int d = dst[k];
    const float ex = __expf(e[k] - m[d]);
    e[k] = ex;
    unsafeAtomicAdd(ssum + d, ex);
}

// ---------------------------------------------------------------------------
// h_N[dst] += (ex/ssum[dst]) * W_h[src].  One wave per edge, 2 cols per lane.
// Gathers and atomic targets are L2-resident (25.6 MB each in 192 MB L2).
// ---------------------------------------------------------------------------
__global__ void k_aggregate(const int* __restrict__ src, const int* __restrict__ dst,
                            const float* __restrict__ ex, const float* __restrict__ ssum,
                            const float* __restrict__ Wh, float* __restrict__ hN) {
    const long long gid = (long long)blockIdx.x * blockDim.x + threadIdx.x;
    const int k    = (int)(gid >> 5);
    const int lane = (int)(gid & 31);
    if (k >= N_EDGES) return;
    const int s = src[k], d = dst[k];
    const float alpha = ex[k] / ssum[d];
    const v2f wv = *(const v2f*)(Wh + (size_t)s * OUT_DIM + 2 * lane);
    float* out = hN + (size_t)d * OUT_DIM + 2 * lane;
    unsafeAtomicAdd(out,     alpha * wv.x);
    unsafeAtomicAdd(out + 1, alpha * wv.y);
}

// ---------------------------------------------------------------------------
extern "C" void kernel_launch(void* const* d_in, const int* in_sizes, int n_in,
                              void* d_out, int out_size, void* d_ws, size_t ws_size,
                              hipStream_t stream) {
    const float* h   = (const float*)d_in[0];
    const int*   src = (const int*)  d_in[1];
    const int*   dst = (const int*)  d_in[2];
    const float* W   = (const float*)d_in[3];
    const float* a   = (const float*)d_in[4];
    float* hN = (float*)d_out;

    // Workspace layout (floats): Wh | s_src | s_dst | e (reused as ex) | m | ssum
    float* ws    = (float*)d_ws;
    float* Wh    = ws;
    float* s_src = Wh    + (size_t)N_NODES * OUT_DIM;
    float* s_dst = s_src + N_NODES;
    float* e     = s_dst + N_NODES;
    float* m     = e     + N_EDGES;
    float* ssum  = m     + N_NODES;

    k_zero<<<1024, 256, 0, stream>>>(hN, m, ssum);
    k_gemm<<<(N_TILES + 7) / 8, 256, 0, stream>>>(h, W, a, Wh, s_src, s_dst);
    k_edge_e<<<(N_EDGES + 255) / 256, 256, 0, stream>>>(src, dst, s_src, s_dst,
                                                        e, (unsigned*)m);
    k_edge_exp<<<(N_EDGES + 255) / 256, 256, 0, stream>>>(dst, m, e, ssum);
    k_aggregate<<<(N_EDGES * 32 + 255) / 256, 256, 0, stream>>>(src, dst, e, ssum,
                                                                Wh, hN);
}